// PerClassBCEFocalLosswithLogits_71554155151524
// MI455X (gfx1250) — compile-verified
//
#include <hip/hip_runtime.h>

#define GAMMA 0.2f
#define ALPHA 0.6f

typedef float v2f __attribute__((ext_vector_type(2)));
typedef float v4f __attribute__((ext_vector_type(4)));
typedef float v8f __attribute__((ext_vector_type(8)));
typedef int   v4i __attribute__((ext_vector_type(4)));

// loss = -a * (1-q)^gamma * log(q), q = sigmoid(y), y = t ? x : -x, a = t ? ALPHA : 1-ALPHA
// log q = -sp, log(1-q) = -y - sp, sp = log(1 + exp(-y))
__device__ __forceinline__ float focal_elem(float x, int t) {
    const bool  pos = (t == 1);
    const float y   = pos ? x : -x;
    const float a   = pos ? ALPHA : (1.0f - ALPHA);
    const float e   = __expf(-y);                 // v_exp_f32
    const float sp  = __logf(1.0f + e);           // v_log_f32 : softplus(-y)
    const float pw  = __expf(GAMMA * (-y - sp));  // (1-q)^gamma
    return a * pw * sp;
}

// Wave32 sum via V_WMMA_F32_16X16X4_F32:
//  A (16x4): lane m (0-15) holds A[m][0]=acc, A[m][1]=0 ; lane m+16 holds A[m][2]=acc, A[m][3]=0
//  B (4x16) = all ones  ->  D[m][n] = acc_m + acc_{m+16}  (independent of n)
//  D VGPR r: lanes 0-15 = D[r][n], lanes 16-31 = D[8+r][n] -> sum 8 VGPRs, then xor-16 add.
__device__ __forceinline__ float wave_sum_wmma(float acc) {
    v2f A; A.x = acc;  A.y = 0.0f;
    v2f B; B.x = 1.0f; B.y = 1.0f;
    v8f C = {};
    C = __builtin_amdgcn_wmma_f32_16x16x4_f32(false, A, false, B, (short)0, C, false, false);
    float h = C[0] + C[1] + C[2] + C[3] + C[4] + C[5] + C[6] + C[7];
    return h + __shfl_xor(h, 16, 32);
}

__global__ void focal_partial_kernel(const float* __restrict__ logits,
                                     const int*   __restrict__ target,
                                     float*       __restrict__ partial,
                                     long long n) {
    const int       tid   = threadIdx.x;
    const long long gsize = (long long)gridDim.x * blockDim.x;
    const long long gtid  = (long long)blockIdx.x * blockDim.x + tid;
    const long long n4    = n >> 2;

    const v4f* __restrict__ l4 = (const v4f*)logits;
    const v4i* __restrict__ t4 = (const v4i*)target;

    float acc = 0.0f;
    for (long long i = gtid; i < n4; i += gsize) {
        v4f x  = __builtin_nontemporal_load(&l4[i]);   // global_load_b128, NT
        v4i tg = __builtin_nontemporal_load(&t4[i]);   // global_load_b128, NT
        acc += focal_elem(x.x, tg.x);
        acc += focal_elem(x.y, tg.y);
        acc += focal_elem(x.z, tg.z);
        acc += focal_elem(x.w, tg.w);
    }
    // scalar tail (n may not be a multiple of 4)
    for (long long i = (n4 << 2) + gtid; i < n; i += gsize)
        acc += focal_elem(logits[i], target[i]);

    // wave32 reduction through the XDL/WMMA pipe (EXEC is all-1s here: no divergence)
    float wsum = wave_sum_wmma(acc);

    __shared__ float sdata[8];
    const int lane = tid & 31;
    const int wave = tid >> 5;
    if (lane == 0) sdata[wave] = wsum;
    __syncthreads();
    if (tid == 0) {
        float b = 0.0f;
        const int nw = blockDim.x >> 5;
        for (int w = 0; w < nw; ++w) b += sdata[w];
        partial[blockIdx.x] = b;
    }
}

__global__ void focal_final_kernel(const float* __restrict__ partial,
                                   float*       __restrict__ out,
                                   int nparts, float scale) {
    float acc = 0.0f;
    for (int i = threadIdx.x; i < nparts; i += blockDim.x)
        acc += partial[i];

    for (int m = 16; m >= 1; m >>= 1)
        acc += __shfl_xor(acc, m, 32);

    __shared__ float sdata[8];
    const int lane = threadIdx.x & 31;
    const int wave = threadIdx.x >> 5;
    if (lane == 0) sdata[wave] = acc;
    __syncthreads();
    if (threadIdx.x == 0) {
        float t = 0.0f;
        const int nw = blockDim.x >> 5;
        for (int w = 0; w < nw; ++w) t += sdata[w];
        out[0] = t * scale;
    }
}

extern "C" void kernel_launch(void* const* d_in, const int* in_sizes, int n_in,
                              void* d_out, int out_size, void* d_ws, size_t ws_size,
                              hipStream_t stream) {
    const float* logits = (const float*)d_in[0];
    const int*   target = (const int*)d_in[1];
    const long long n   = (long long)in_sizes[0];   // N*C = 41,943,040

    float* partial = (float*)d_ws;

    int blocks = 4096;  // 1M threads -> ~10 x 128-bit loads per thread, saturates HBM
    const size_t maxb = ws_size / sizeof(float);
    if ((size_t)blocks > maxb) blocks = (int)maxb;
    if (blocks < 1) blocks = 1;

    focal_partial_kernel<<<blocks, 256, 0, stream>>>(logits, target, partial, n);
    focal_final_kernel<<<1, 256, 0, stream>>>(partial, (float*)d_out, blocks,
                                              1.0f / (float)n);
}